// Model_28449863368846
// MI455X (gfx1250) — compile-verified
//
#include <hip/hip_runtime.h>

typedef __attribute__((ext_vector_type(2))) float v2f;
typedef __attribute__((ext_vector_type(8))) float v8f;

#define W_  1024
#define H_  1024
#define HW_ (1024 * 1024)

// One block (256 threads = 8 waves) per (b,h) row.
// Each wave processes 8 tiles of 16 pixels with V_WMMA_F32_16X16X4_F32:
//   D(16x16) = A(16x4: [wq;wk;wv;0]) x B(4x16: x-tile) + C(bias)
// lanes 0-15 get q(d0..3)/k(d4..7); lanes 16-31 get v(d0..3), zeros(d4..7).
__global__ __launch_bounds__(256) void qkv_attn_row_kernel(
    const float* __restrict__ x,
    const float* __restrict__ wq, const float* __restrict__ bq,
    const float* __restrict__ wk, const float* __restrict__ bk,
    const float* __restrict__ wv, const float* __restrict__ bv,
    float* __restrict__ out)
{
    __shared__ float s_v[4 * W_];      // v row: [4][1024]
    __shared__ float s_scores[16];     // q.k^T accumulators (c*4+d)

    const int tid  = threadIdx.x;
    const int lane = tid & 31;
    const int wave = tid >> 5;

    const int r = blockIdx.x;          // r = b*H + h
    const int b = r >> 10;
    const int h = r & (H_ - 1);

    const size_t xbase = (size_t)b * 4 * HW_ + (size_t)h * W_;
    const size_t obase = (size_t)r * 4 * W_;

    if (tid < 16) s_scores[tid] = 0.0f;

    // ---- A operand (16x4 f32): rows 0-3 wq, 4-7 wk, 8-11 wv, 12-15 zero.
    // Layout: lanes 0-15 hold K=0 (vgpr0) / K=1 (vgpr1); lanes 16-31 K=2/K=3.
    const int m  = lane & 15;
    const int k0 = (lane < 16) ? 0 : 2;
    float a0, a1;
    if (m < 4)       { a0 = wq[m * 4 + k0];       a1 = wq[m * 4 + k0 + 1]; }
    else if (m < 8)  { a0 = wk[(m - 4) * 4 + k0]; a1 = wk[(m - 4) * 4 + k0 + 1]; }
    else if (m < 12) { a0 = wv[(m - 8) * 4 + k0]; a1 = wv[(m - 8) * 4 + k0 + 1]; }
    else             { a0 = 0.0f;                 a1 = 0.0f; }
    v2f A; A.x = a0; A.y = a1;

    // ---- C operand (bias): vgpr rr = row rr (lanes 0-15) / row rr+8 (lanes 16-31)
    v8f Cm;
    #pragma unroll
    for (int rr = 0; rr < 8; ++rr) {
        float cv;
        if (lane < 16) cv = (rr < 4) ? bq[rr] : bk[rr - 4];   // rows 0-7
        else           cv = (rr < 4) ? bv[rr] : 0.0f;          // rows 8-15
        Cm[rr] = cv;
    }

    __syncthreads();  // s_scores init visible

    float part[16];
    #pragma unroll
    for (int i = 0; i < 16; ++i) part[i] = 0.0f;

    const int col = lane & 15;
    const int cB  = (lane < 16) ? 0 : 2;  // B layout mirrors A K-striping

    // ---- Batch all B-operand loads for this wave's 8 tiles: one 16-load
    // clause, single wait, 16 outstanding requests per wave (MLP).
    const float* pxl = x + xbase + (size_t)cB * HW_ + (size_t)(wave * 128 + col);
    float bxv[8], byv[8];
    #pragma unroll
    for (int t = 0; t < 8; ++t) {
        bxv[t] = pxl[t * 16];
        byv[t] = pxl[HW_ + t * 16];
    }

    #pragma unroll
    for (int t = 0; t < 8; ++t) {
        v2f Bm; Bm.x = bxv[t]; Bm.y = byv[t];

        v8f D = __builtin_amdgcn_wmma_f32_16x16x4_f32(
            false, A, false, Bm, (short)0, Cm, false, false);

        // scores partials: lanes<16 do q*k; lanes>=16 have D[4..7]==0 -> add 0
        #pragma unroll
        for (int c = 0; c < 4; ++c)
            #pragma unroll
            for (int dd = 0; dd < 4; ++dd)
                part[c * 4 + dd] = fmaf(D[c], D[4 + dd], part[c * 4 + dd]);

        // stash v (rows 8-11 live in lanes 16-31, vgprs 0-3)
        if (lane >= 16) {
            const int cc = wave * 128 + t * 16 + col;
            s_v[0 * W_ + cc] = D[0];
            s_v[1 * W_ + cc] = D[1];
            s_v[2 * W_ + cc] = D[2];
            s_v[3 * W_ + cc] = D[3];
        }
    }

    // wave32 butterfly reduction of the 16 score partials
    #pragma unroll
    for (int i = 0; i < 16; ++i) {
        float p = part[i];
        #pragma unroll
        for (int off = 16; off >= 1; off >>= 1)
            p += __shfl_xor(p, off, 32);
        part[i] = p;
    }
    if (lane == 0) {
        #pragma unroll
        for (int i = 0; i < 16; ++i)
            atomicAdd(&s_scores[i], part[i]);   // ds_add_f32
    }

    __syncthreads();  // scores + s_v complete

    // softmax over d, scale = 1/sqrt(W) = 0.03125 (computed redundantly; tiny)
    float attn[16];
    #pragma unroll
    for (int c = 0; c < 4; ++c) {
        float s0 = s_scores[c * 4 + 0] * 0.03125f;
        float s1 = s_scores[c * 4 + 1] * 0.03125f;
        float s2 = s_scores[c * 4 + 2] * 0.03125f;
        float s3 = s_scores[c * 4 + 3] * 0.03125f;
        float mx = fmaxf(fmaxf(s0, s1), fmaxf(s2, s3));
        float e0 = __expf(s0 - mx), e1 = __expf(s1 - mx);
        float e2 = __expf(s2 - mx), e3 = __expf(s3 - mx);
        float inv = 1.0f / (e0 + e1 + e2 + e3);
        attn[c * 4 + 0] = e0 * inv; attn[c * 4 + 1] = e1 * inv;
        attn[c * 4 + 2] = e2 * inv; attn[c * 4 + 3] = e3 * inv;
    }

    // pass 2: out[c][w] = sum_d attn[c][d] * v[d][w]; 4 columns per thread
    const int wcol = tid * 4;
    const float4 v0 = *(const float4*)&s_v[0 * W_ + wcol];
    const float4 v1 = *(const float4*)&s_v[1 * W_ + wcol];
    const float4 v2 = *(const float4*)&s_v[2 * W_ + wcol];
    const float4 v3 = *(const float4*)&s_v[3 * W_ + wcol];
    #pragma unroll
    for (int c = 0; c < 4; ++c) {
        const float a0c = attn[c * 4 + 0], a1c = attn[c * 4 + 1];
        const float a2c = attn[c * 4 + 2], a3c = attn[c * 4 + 3];
        float4 o;
        o.x = a0c * v0.x + a1c * v1.x + a2c * v2.x + a3c * v3.x;
        o.y = a0c * v0.y + a1c * v1.y + a2c * v2.y + a3c * v3.y;
        o.z = a0c * v0.z + a1c * v1.z + a2c * v2.z + a3c * v3.z;
        o.w = a0c * v0.w + a1c * v1.w + a2c * v2.w + a3c * v3.w;
        *(float4*)&out[obase + (size_t)c * W_ + wcol] = o;
    }
}

extern "C" void kernel_launch(void* const* d_in, const int* in_sizes, int n_in,
                              void* d_out, int out_size, void* d_ws, size_t ws_size,
                              hipStream_t stream) {
    const float* x  = (const float*)d_in[0];
    const float* wq = (const float*)d_in[1];
    const float* bq = (const float*)d_in[2];
    const float* wk = (const float*)d_in[3];
    const float* bk = (const float*)d_in[4];
    const float* wv = (const float*)d_in[5];
    const float* bv = (const float*)d_in[6];
    float* out = (float*)d_out;

    dim3 grid(16 * 1024);   // B * H rows
    dim3 block(256);        // 8 wave32s
    qkv_attn_row_kernel<<<grid, block, 0, stream>>>(x, wq, bq, wk, bk, wv, bv, out);
}